// SAGE_15925738733668
// MI455X (gfx1250) — compile-verified
//
#include <hip/hip_runtime.h>

#define N_NODES 50000
#define N_EDGES 800000
#define IN_F    128
#define H_F     128
#define N_CLS   64
#define M_TILES (N_NODES / 16)   // 3125, exact

typedef float v2f __attribute__((ext_vector_type(2)));
typedef float v8f __attribute__((ext_vector_type(8)));

// ---------------------------------------------------------------------------
// Wc = Wr0 + Wskip0   (fold skip projection into the self-term weight)
// ---------------------------------------------------------------------------
__global__ void sage_wadd_kernel(const float* __restrict__ a,
                                 const float* __restrict__ b,
                                 float* __restrict__ out, int n) {
    int i = blockIdx.x * blockDim.x + threadIdx.x;
    if (i < n) out[i] = a[i] + b[i];
}

// ---------------------------------------------------------------------------
// degree count (float) per destination node
// ---------------------------------------------------------------------------
__global__ void sage_degree_kernel(const int* __restrict__ dst,
                                   float* __restrict__ cnt, int nE) {
    int e = blockIdx.x * blockDim.x + threadIdx.x;
    if (e < nE) atomicAdd(&cnt[dst[e]], 1.0f);
}

// cnt[i] -> 1 / max(cnt[i], 1)
__global__ void sage_invcnt_kernel(float* __restrict__ cnt, int n) {
    int i = blockIdx.x * blockDim.x + threadIdx.x;
    if (i < n) cnt[i] = 1.0f / fmaxf(cnt[i], 1.0f);
}

// ---------------------------------------------------------------------------
// scatter-add: agg[dst] += X[src]   (one wave32 per edge, float4 per lane,
// F = 128 floats per row -> 32 lanes * 4 floats)
// ---------------------------------------------------------------------------
__global__ __launch_bounds__(256)
void sage_scatter_kernel(const float* __restrict__ X,
                         const int* __restrict__ src,
                         const int* __restrict__ dst,
                         float* __restrict__ agg, int nE) {
    long long t = (long long)blockIdx.x * blockDim.x + threadIdx.x;
    int e = (int)(t >> 5);         // edge id (uniform across the wave)
    int c = ((int)t & 31) << 2;    // float offset within the 128-float row
    if (e >= nE) return;
    int s = src[e];
    int d = dst[e];
    const float4 v = *(const float4*)(X + (size_t)s * 128 + c);
    float* ap = agg + (size_t)d * 128 + c;
    atomicAdd(ap + 0, v.x);
    atomicAdd(ap + 1, v.y);
    atomicAdd(ap + 2, v.z);
    atomicAdd(ap + 3, v.w);
}

// ---------------------------------------------------------------------------
// Fused dual-GEMM SAGE layer with fp32 WMMA:
//   Y = act( (Agg * invdeg) @ Wl^T  +  Xs @ Wr^T  +  bias )
// One wave per 16x16 output tile; K stepped by 4 with V_WMMA_F32_16X16X4_F32,
// both products chained through the same accumulator.
// A layout (16x4 f32): lane = M (mod 16); VGPR pair holds K = kb, kb+1 with
//   kb = (lane>=16 ? 2 : 0).  B mirrors with lane = N.
// C/D layout: VGPR j -> row j (lanes 0-15) / row j+8 (lanes 16-31), lane = N.
// ---------------------------------------------------------------------------
template <bool RELU>
__global__ __launch_bounds__(256)
void sage_gemm_wmma(const float* __restrict__ Agg,    // [M,K] summed neighbors
                    const float* __restrict__ invdeg, // [M]   1/deg
                    const float* __restrict__ Wl,     // [Nout,K]
                    const float* __restrict__ Xs,     // [M,K] self features
                    const float* __restrict__ Wr,     // [Nout,K]
                    const float* __restrict__ bias,   // [Nout]
                    float* __restrict__ Y,            // [M,Nout]
                    int K, int Nout, int nMTiles) {
    const int lane  = threadIdx.x & 31;
    const int wave  = threadIdx.x >> 5;
    const int mTile = blockIdx.x * 8 + wave;
    if (mTile >= nMTiles) return;              // wave-uniform: EXEC stays all-1
    const int nTile = blockIdx.y;

    const int l15 = lane & 15;
    const int m   = mTile * 16 + l15;          // A row for this lane
    const int col = nTile * 16 + l15;          // B column / output column
    const int kb  = (lane >> 4) << 1;          // 0 or 2

    const float* a1p = Agg + (size_t)m * K;
    const float* a2p = Xs  + (size_t)m * K;
    const float* w1p = Wl  + (size_t)col * K;
    const float* w2p = Wr  + (size_t)col * K;
    const float  inv = invdeg[m];

    __builtin_prefetch(a1p, 0, 0);             // global_prefetch_b8
    __builtin_prefetch(a2p, 0, 0);

    v8f c = {};
#pragma unroll 4
    for (int k = 0; k < K; k += 4) {
        v2f a1 = *(const v2f*)(a1p + k + kb);
        a1 *= inv;                                          // fold mean divide
        v2f b1 = *(const v2f*)(w1p + k + kb);
        c = __builtin_amdgcn_wmma_f32_16x16x4_f32(false, a1, false, b1,
                                                  (short)0, c, false, false);
        v2f a2 = *(const v2f*)(a2p + k + kb);
        v2f b2 = *(const v2f*)(w2p + k + kb);
        c = __builtin_amdgcn_wmma_f32_16x16x4_f32(false, a2, false, b2,
                                                  (short)0, c, false, false);
    }

    const float bv      = bias[col];
    const int   rowBase = mTile * 16 + ((lane >> 4) << 3);
#pragma unroll
    for (int j = 0; j < 8; ++j) {
        float v = c[j] + bv;
        if (RELU) v = fmaxf(v, 0.0f);
        Y[(size_t)(rowBase + j) * Nout + col] = v;
    }
}

// ---------------------------------------------------------------------------
// launch
// ---------------------------------------------------------------------------
extern "C" void kernel_launch(void* const* d_in, const int* in_sizes, int n_in,
                              void* d_out, int out_size, void* d_ws, size_t ws_size,
                              hipStream_t stream) {
    const float* x      = (const float*)d_in[0];
    const int*   eidx   = (const int*)d_in[1];
    const float* Wl0    = (const float*)d_in[2];
    const float* bl0    = (const float*)d_in[3];
    const float* Wr0    = (const float*)d_in[4];
    const float* Wskip0 = (const float*)d_in[5];
    const float* Wl1    = (const float*)d_in[6];
    const float* bl1    = (const float*)d_in[7];
    const float* Wr1    = (const float*)d_in[8];
    float*       out    = (float*)d_out;

    const int* src = eidx;             // edge_index[0]
    const int* dst = eidx + N_EDGES;   // edge_index[1]

    // workspace layout
    char*  ws   = (char*)d_ws;
    float* cnt  = (float*)(ws);                                   // 200 KB (-> invdeg)
    float* Wc   = (float*)(ws + 204800);                          // 64 KB
    float* agg  = (float*)(ws + 204800 + 65536);                  // 25.6 MB (both layers)
    float* h    = (float*)(ws + 204800 + 65536 + 25600000);       // 25.6 MB

    const size_t aggBytes = (size_t)N_NODES * IN_F * sizeof(float);

    // ---- shared precompute -------------------------------------------------
    hipMemsetAsync(cnt, 0, N_NODES * sizeof(float), stream);
    hipMemsetAsync(agg, 0, aggBytes, stream);
    sage_wadd_kernel<<<(IN_F * IN_F + 255) / 256, 256, 0, stream>>>(
        Wr0, Wskip0, Wc, IN_F * IN_F);
    sage_degree_kernel<<<(N_EDGES + 255) / 256, 256, 0, stream>>>(dst, cnt, N_EDGES);
    sage_invcnt_kernel<<<(N_NODES + 255) / 256, 256, 0, stream>>>(cnt, N_NODES);

    // ---- layer 0: h = relu(mean(x)@Wl0^T + x@(Wr0+Wskip0)^T + bl0) ---------
    {
        int nThreads = N_EDGES * 32;
        sage_scatter_kernel<<<(nThreads + 255) / 256, 256, 0, stream>>>(
            x, src, dst, agg, N_EDGES);
        dim3 grid((M_TILES + 7) / 8, H_F / 16);
        sage_gemm_wmma<true><<<grid, 256, 0, stream>>>(
            agg, cnt, Wl0, x, Wc, bl0, h, IN_F, H_F, M_TILES);
    }

    // ---- layer 1: out = mean(h)@Wl1^T + h@Wr1^T + bl1 ----------------------
    {
        hipMemsetAsync(agg, 0, aggBytes, stream);
        int nThreads = N_EDGES * 32;
        sage_scatter_kernel<<<(nThreads + 255) / 256, 256, 0, stream>>>(
            h, src, dst, agg, N_EDGES);
        dim3 grid((M_TILES + 7) / 8, N_CLS / 16);
        sage_gemm_wmma<false><<<grid, 256, 0, stream>>>(
            agg, cnt, Wl1, h, Wr1, bl1, out, H_F, N_CLS, M_TILES);
    }
}